// EquivairantMultiheadAttention_29798483100008
// MI455X (gfx1250) — compile-verified
//
#include <hip/hip_runtime.h>
#include <hip/hip_bf16.h>

// Problem constants (from reference setup_inputs)
#define BSZ 4
#define NSEQ 512
#define CDIM 256
#define HEADS 8
#define DHEAD 32
#define DG 6
#define HID 16

typedef float v2f __attribute__((ext_vector_type(2)));
typedef float v8f __attribute__((ext_vector_type(8)));
typedef unsigned int u32x4 __attribute__((ext_vector_type(4)));
typedef int i32x4 __attribute__((ext_vector_type(4)));
typedef int i32x8 __attribute__((ext_vector_type(8)));

__device__ __forceinline__ v8f wmma_f32(v2f a, v2f b, v8f c) {
  // D = A(16x4 f32) * B(4x16 f32) + C(16x16 f32)
  return __builtin_amdgcn_wmma_f32_16x16x4_f32(false, a, false, b, (short)0, c,
                                               false, false);
}

__device__ __forceinline__ float swish_f(float x) {
  return x / (1.0f + __expf(-x));
}

// -------------------------------------------------------------------------
// Y[R x 256] = X[R x 256] @ W[256 x 256] + bias ; one wave per 16x16 tile.
// grid = (R/16, 256/16), block = 32
// -------------------------------------------------------------------------
__global__ __launch_bounds__(32) void proj_kernel(
    const float* __restrict__ X, const float* __restrict__ W,
    const float* __restrict__ bias, float* __restrict__ Y) {
  const int l = threadIdx.x;
  const int half = l >> 4;
  const int lm = l & 15;
  const int row0 = blockIdx.x * 16;
  const int col0 = blockIdx.y * 16;
  const int row = row0 + lm;
  const int col = col0 + lm;

  __builtin_prefetch(X + (size_t)row * CDIM, 0, 3);  // global_prefetch_b8
  __builtin_prefetch(W + col, 0, 3);

  v8f acc = {};
#pragma unroll 4
  for (int k0 = 0; k0 < CDIM; k0 += 4) {
    const int kk = k0 + 2 * half;                       // K slice per half-wave
    v2f a = *(const v2f*)(X + (size_t)row * CDIM + kk); // A: M=lane, K in vgprs
    v2f b;                                              // B: N=lane, K in vgprs
    b.x = W[(size_t)kk * CDIM + col];
    b.y = W[(size_t)(kk + 1) * CDIM + col];
    acc = wmma_f32(a, b, acc);
  }
  const float bv = bias[col];
#pragma unroll
  for (int r = 0; r < 8; ++r) {
    Y[(size_t)(row0 + r + 8 * half) * CDIM + col] = acc[r] + bv;
  }
}

// -------------------------------------------------------------------------
// A_feat: pre[b,h,n,m] = (Q[b,n,h,:] . K[b,m,h,:]) / sqrt(C)
// grid = (N/16, N/16, B*H), block = 32
// -------------------------------------------------------------------------
__global__ __launch_bounds__(32) void feat_kernel(const float* __restrict__ Q,
                                                  const float* __restrict__ Km,
                                                  float* __restrict__ pre) {
  const int l = threadIdx.x;
  const int half = l >> 4;
  const int lm = l & 15;
  const int n0 = blockIdx.x * 16;
  const int m0 = blockIdx.y * 16;
  const int bh = blockIdx.z;
  const int bb = bh >> 3;
  const int h = bh & 7;

  const float* Qb = Q + ((size_t)(bb * NSEQ) + n0) * CDIM + h * DHEAD;
  const float* Kb = Km + ((size_t)(bb * NSEQ) + m0) * CDIM + h * DHEAD;

  v8f acc = {};
#pragma unroll
  for (int k0 = 0; k0 < DHEAD; k0 += 4) {
    const int kk = k0 + 2 * half;
    v2f a = *(const v2f*)(Qb + (size_t)lm * CDIM + kk); // rows of Q
    v2f b = *(const v2f*)(Kb + (size_t)lm * CDIM + kk); // rows of K == cols of K^T
    acc = wmma_f32(a, b, acc);
  }
  const float scale = 0.0625f; // 1/sqrt(256)
  float* prow = pre + ((size_t)bh * NSEQ + n0) * NSEQ + m0;
#pragma unroll
  for (int r = 0; r < 8; ++r)
    prow[(size_t)(r + 8 * half) * NSEQ + lm] = acc[r] * scale;
}

// -------------------------------------------------------------------------
// A_loc MLP (6->16->16->1, swish), add to pre, apply mask.
// grid = (N/16 m-tiles, N, B), block = (32, 8): wave y == head.
// The shared 16x6 g tile is DMA'd to LDS once per block via the Tensor Data
// Mover (TDM), instead of 8 waves issuing identical global loads.
// -------------------------------------------------------------------------
__global__ __launch_bounds__(256) void loc_kernel(
    const float* __restrict__ g, const float* __restrict__ w1,
    const float* __restrict__ b1, const float* __restrict__ w2,
    const float* __restrict__ b2, const float* __restrict__ w3,
    const float* __restrict__ b3, const int* __restrict__ mask,
    float* __restrict__ pre) {
  __shared__ __align__(16) float gs[16 * DG + 8]; // TDM-staged g tile (96 f32)
  __shared__ float xt[HEADS][16 * 17]; // stride-17 pad: conflict-free A reads
  const int l = threadIdx.x;
  const int h = threadIdx.y;
  const int half = l >> 4;
  const int lm = l & 15;
  const int m0 = blockIdx.x * 16;
  const int nn = blockIdx.y;
  const int bb = blockIdx.z;

  const float* gbase = g + (((size_t)(bb * NSEQ) + nn) * NSEQ + m0) * DG;

  if (h == 0) {
    // Tensor DMA descriptor (D#) per CDNA5 ISA:
    // 1-D tile of 96 f32 elements, global -> LDS.
    const uint32_t lds_off = (uint32_t)(uintptr_t)(&gs[0]); // low 32b = LDS addr
    const uint64_t ga = (uint64_t)(uintptr_t)gbase;
    u32x4 g0;
    g0[0] = 1u;                  // count=1 (valid), load, user mode
    g0[1] = lds_off;             // lds_addr (bytes)
    g0[2] = (uint32_t)ga;        // global_addr[31:0]
    g0[3] = (uint32_t)((ga >> 32) & 0x01FFFFFFu) | (2u << 30); // [56:32] | type=2
    i32x8 g1;
    g1[0] = 2 << 16;             // workgroup_mask=0, data_size=2 (4 bytes)
    g1[1] = 96 << 16;            // tensor_dim0 = 96 (low 16 bits)
    g1[2] = 1 << 16;             // tensor_dim0 hi = 0, tensor_dim1 = 1
    g1[3] = 96 << 16;            // tensor_dim1 hi = 0, tile_dim0 = 96
    g1[4] = 1;                   // tile_dim1 = 1, tile_dim2 = 0
    g1[5] = 96;                  // tensor_dim0_stride = 96
    g1[6] = 0;                   // stride hi / tensor_dim1_stride lo
    g1[7] = 0;                   // tensor_dim1_stride hi
    i32x4 z4 = {0, 0, 0, 0};     // groups 2/3 unused (<=2D tensor)
    i32x8 z8 = {0, 0, 0, 0, 0, 0, 0, 0};
    __builtin_amdgcn_tensor_load_to_lds(g0, g1, z4, z4, z8, 0);
    __builtin_amdgcn_s_wait_tensorcnt(0); // TDM data landed in LDS
  }
  __syncthreads();

  // ---- layer 1: A = g (16 pairs x 6, padded to K=8), B = w1[h] (6x16) ----
  v8f acc = {};
  {
    const int k0 = 2 * half; // half 0 -> K 0,1 ; half 1 -> K 2,3
    v2f a = *(const v2f*)(&gs[lm * DG + k0]);
    v2f wb;
    wb.x = w1[(size_t)(h * DG + k0) * HID + lm];
    wb.y = w1[(size_t)(h * DG + k0 + 1) * HID + lm];
    acc = wmma_f32(a, wb, acc);

    v2f a2 = {0.f, 0.f}, wb2 = {0.f, 0.f}; // half 0 -> K 4,5 ; half 1 -> pad 6,7
    if (half == 0) {
      a2.x = gs[lm * DG + 4];
      a2.y = gs[lm * DG + 5];
      wb2.x = w1[(size_t)(h * DG + 4) * HID + lm];
      wb2.y = w1[(size_t)(h * DG + 5) * HID + lm];
    }
    acc = wmma_f32(a2, wb2, acc); // full EXEC at the WMMA itself
  }

  // swish + transpose D-layout -> A-layout via LDS
  const float bias1 = b1[h * HID + lm];
  float* xw = &xt[h][0];
#pragma unroll
  for (int r = 0; r < 8; ++r)
    xw[(r + 8 * half) * 17 + lm] = swish_f(acc[r] + bias1);

  // ---- layer 2: A = x1 (16x16 from LDS), B = w2[h] (16x16) ----
  v8f acc2 = {};
#pragma unroll
  for (int ks = 0; ks < 4; ++ks) {
    const int kk = ks * 4 + 2 * half;
    v2f a;
    a.x = xw[lm * 17 + kk];
    a.y = xw[lm * 17 + kk + 1];
    v2f wb;
    wb.x = w2[(size_t)(h * HID + kk) * HID + lm];
    wb.y = w2[(size_t)(h * HID + kk + 1) * HID + lm];
    acc2 = wmma_f32(a, wb, acc2);
  }

  // ---- layer 3: dot over hid via cross-lane reduction within each half ----
  const float bias2 = b2[h * HID + lm];
  const float w3v = w3[h * HID + lm];
  float partial[8];
#pragma unroll
  for (int r = 0; r < 8; ++r)
    partial[r] = swish_f(acc2[r] + bias2) * w3v;
#pragma unroll
  for (int off = 1; off < 16; off <<= 1) {
#pragma unroll
    for (int r = 0; r < 8; ++r)
      partial[r] += __shfl_xor(partial[r], off, 32);
  }

  const float b3v = b3[h];
  if (lm < 8) { // lanes 0..7 of each half commit 8 rows each
    const int r = lm;
    const int m = m0 + r + 8 * half;
    const float aloc = swish_f(partial[r] + b3v);
    const size_t idx =
        (((size_t)(bb * HEADS + h) * NSEQ) + nn) * (size_t)NSEQ + m;
    float val = aloc + pre[idx];
    val = (mask[bb * NSEQ + m] != 0) ? val : -1e38f;
    pre[idx] = val;
  }
}

// -------------------------------------------------------------------------
// Row softmax over m (row length 512). One wave per (b,h,n) row.
// -------------------------------------------------------------------------
__global__ __launch_bounds__(32) void softmax_kernel(float* __restrict__ pre) {
  const size_t row = blockIdx.x;
  float* p = pre + row * (size_t)NSEQ;
  const int l = threadIdx.x;

  float vals[16];
  float mx = -3.4e38f;
#pragma unroll
  for (int i = 0; i < 16; ++i) {
    vals[i] = p[l + i * 32];
    mx = fmaxf(mx, vals[i]);
  }
#pragma unroll
  for (int off = 1; off < 32; off <<= 1) mx = fmaxf(mx, __shfl_xor(mx, off, 32));
  float sum = 0.f;
#pragma unroll
  for (int i = 0; i < 16; ++i) {
    vals[i] = __expf(vals[i] - mx);
    sum += vals[i];
  }
#pragma unroll
  for (int off = 1; off < 32; off <<= 1) sum += __shfl_xor(sum, off, 32);
  const float inv = 1.0f / sum;
#pragma unroll
  for (int i = 0; i < 16; ++i) p[l + i * 32] = vals[i] * inv;
}

// -------------------------------------------------------------------------
// ctx[b,n,h*32+dd] = sum_m att[b,h,n,m] * V[b,m,h*32+dd]
// grid = (N/16, DHEAD/16, B*H), block = 32; K = 512 -> 128 WMMAs per tile.
// -------------------------------------------------------------------------
__global__ __launch_bounds__(32) void av_kernel(const float* __restrict__ att,
                                                const float* __restrict__ V,
                                                float* __restrict__ ctx) {
  const int l = threadIdx.x;
  const int half = l >> 4;
  const int lm = l & 15;
  const int n0 = blockIdx.x * 16;
  const int d0 = blockIdx.y * 16;
  const int bh = blockIdx.z;
  const int bb = bh >> 3;
  const int h = bh & 7;

  const float* arow = att + (((size_t)bh * NSEQ) + n0 + lm) * (size_t)NSEQ;
  const float* vcol = V + (size_t)bb * NSEQ * CDIM + h * DHEAD + d0 + lm;

  __builtin_prefetch(arow, 0, 3);
  __builtin_prefetch(vcol, 0, 3);

  v8f acc = {};
#pragma unroll 4
  for (int k0 = 0; k0 < NSEQ; k0 += 4) {
    const int kk = k0 + 2 * half;
    v2f a = *(const v2f*)(arow + kk);
    v2f b;
    b.x = vcol[(size_t)kk * CDIM];
    b.y = vcol[(size_t)(kk + 1) * CDIM];
    acc = wmma_f32(a, b, acc);
  }
  float* crow = ctx + ((size_t)(bb * NSEQ) + n0) * CDIM + h * DHEAD + d0;
#pragma unroll
  for (int r = 0; r < 8; ++r)
    crow[(size_t)(r + 8 * half) * CDIM + lm] = acc[r];
}

// -------------------------------------------------------------------------
extern "C" void kernel_launch(void* const* d_in, const int* in_sizes, int n_in,
                              void* d_out, int out_size, void* d_ws,
                              size_t ws_size, hipStream_t stream) {
  const float* pairwise_g = (const float*)d_in[0];
  const float* coset      = (const float*)d_in[1];
  const float* fc_q_w     = (const float*)d_in[2];
  const float* fc_q_b     = (const float*)d_in[3];
  const float* fc_k_w     = (const float*)d_in[4];
  const float* fc_k_b     = (const float*)d_in[5];
  const float* in_w       = (const float*)d_in[6];
  const float* in_b       = (const float*)d_in[7];
  const float* out_w      = (const float*)d_in[8];
  const float* out_b      = (const float*)d_in[9];
  const float* loc_w1     = (const float*)d_in[10];
  const float* loc_b1     = (const float*)d_in[11];
  const float* loc_w2     = (const float*)d_in[12];
  const float* loc_b2     = (const float*)d_in[13];
  const float* loc_w3     = (const float*)d_in[14];
  const float* loc_b3     = (const float*)d_in[15];
  const int*   mask       = (const int*)d_in[16];
  float* out = (float*)d_out;

  // workspace layout (floats)
  float* ws = (float*)d_ws;
  const size_t BN = (size_t)BSZ * NSEQ;            // 2048
  float* Q   = ws;                                  // 2048*256
  float* K   = Q + BN * CDIM;
  float* V   = K + BN * CDIM;
  float* ctx = V + BN * CDIM;
  float* pre = ctx + BN * CDIM;                     // B*H*N*N floats (32 MB)

  const dim3 wave(32, 1, 1);
  const dim3 projGrid(BN / 16, CDIM / 16, 1);

  // Q / K / V projections
  proj_kernel<<<projGrid, wave, 0, stream>>>(coset, fc_q_w, fc_q_b, Q);
  proj_kernel<<<projGrid, wave, 0, stream>>>(coset, fc_k_w, fc_k_b, K);
  proj_kernel<<<projGrid, wave, 0, stream>>>(coset, in_w, in_b, V);

  // A_feat logits
  feat_kernel<<<dim3(NSEQ / 16, NSEQ / 16, BSZ * HEADS), wave, 0, stream>>>(
      Q, K, pre);

  // A_loc MLP + mask, fused into logits (g tile staged via TDM)
  loc_kernel<<<dim3(NSEQ / 16, NSEQ, BSZ), dim3(32, HEADS), 0, stream>>>(
      pairwise_g, loc_w1, loc_b1, loc_w2, loc_b2, loc_w3, loc_b3, mask, pre);

  // softmax over m
  softmax_kernel<<<dim3(BSZ * HEADS * NSEQ), wave, 0, stream>>>(pre);

  // att @ V
  av_kernel<<<dim3(NSEQ / 16, DHEAD / 16, BSZ * HEADS), wave, 0, stream>>>(
      pre, V, ctx);

  // output projection
  proj_kernel<<<projGrid, wave, 0, stream>>>(ctx, out_w, out_b, out);
}